// FPSBallQueryModule_81183471829352
// MI455X (gfx1250) — compile-verified
//
#include <hip/hip_runtime.h>
#include <cstdint>
#include <cstddef>

#define B_  16
#define N_  16384
#define M_  1024
#define NS_ 32

static constexpr float R2_ = 0.01f; // RADIUS^2 = 0.1^2

typedef __attribute__((ext_vector_type(2))) float v2f;
typedef __attribute__((ext_vector_type(8))) float v8f;

// ===================== Kernel 1: furthest point sampling =====================
// One block per batch, 1024 threads (32 wave32s), 16 points/thread in regs.
// Sequential over M=1024 selections; argmax via shfl tree + LDS cross-wave.
__global__ __launch_bounds__(1024)
void fps_kernel(const float* __restrict__ xyz, float* __restrict__ cent)
{
  const int b    = blockIdx.x;
  const int t    = threadIdx.x;
  const int lane = t & 31;
  const int wv   = t >> 5;
  const float* __restrict__ p = xyz + (size_t)b * (N_ * 3);

  float px[16], py[16], pz[16], md[16];
#pragma unroll
  for (int j = 0; j < 16; ++j) {
    const int i = (j << 10) + t;          // coalesced across lanes
    px[j] = p[3 * i + 0];
    py[j] = p[3 * i + 1];
    pz[j] = p[3 * i + 2];
    md[j] = 1e10f;
  }

  __shared__ float s_best[32];
  __shared__ int   s_bidx[32];
  __shared__ int   s_far;

  int far = 0;
  for (int m = 0; m < M_; ++m) {
    // broadcast load of current centroid (uniform address -> one line)
    const float cx = p[3 * far + 0];
    const float cy = p[3 * far + 1];
    const float cz = p[3 * far + 2];
    if (t == 0) {
      float* c = cent + ((size_t)b * M_ + m) * 3;
      c[0] = cx; c[1] = cy; c[2] = cz;    // centroid output = xyz[fps_idx[m]]
    }
    float best = -1.0f; int bidx = 0x7fffffff;
#pragma unroll
    for (int j = 0; j < 16; ++j) {
      const float dx = px[j] - cx, dy = py[j] - cy, dz = pz[j] - cz;
      const float d  = dx * dx + dy * dy + dz * dz;
      const float nm = fminf(md[j], d);
      md[j] = nm;
      const int i = (j << 10) + t;
      if (nm > best || (nm == best && i < bidx)) { best = nm; bidx = i; }
    }
    // wave32 argmax (lowest index wins ties, matching jnp.argmax)
#pragma unroll
    for (int o = 16; o > 0; o >>= 1) {
      const float ob = __shfl_down(best, o, 32);
      const int   oi = __shfl_down(bidx, o, 32);
      if (ob > best || (ob == best && oi < bidx)) { best = ob; bidx = oi; }
    }
    if (lane == 0) { s_best[wv] = best; s_bidx[wv] = bidx; }
    __syncthreads();
    if (wv == 0) {
      float bb = s_best[lane]; int bi = s_bidx[lane];
#pragma unroll
      for (int o = 16; o > 0; o >>= 1) {
        const float ob = __shfl_down(bb, o, 32);
        const int   oi = __shfl_down(bi, o, 32);
        if (ob > bb || (ob == bb && oi < bi)) { bb = ob; bi = oi; }
      }
      if (lane == 0) s_far = bi;
    }
    __syncthreads();
    far = s_far;
  }
}

// ================= Kernel 2: ball query + grouping via WMMA ==================
// Block = 256 threads (8 waves), handles 128 centroids of one batch.
// Whole batch point cloud (192KB SoA) staged into LDS with async loads.
// dist2 tile [16 centroids x 16 points] = one V_WMMA_F32_16X16X4_F32:
//   A row m = (-2cx, -2cy, -2cz, |c|^2), B col n = (px, py, pz, 1), C = |p|^2.
__global__ __launch_bounds__(256)
void ballq_kernel(const float* __restrict__ xyz, const float* __restrict__ cent,
                  float* __restrict__ grouped)
{
  extern __shared__ float smem[];         // sx[N_], sy[N_], sz[N_] = 192KB
  float* sx = smem;
  float* sy = smem + N_;
  float* sz = smem + 2 * N_;

  const int b     = blockIdx.x >> 3;      // 8 centroid tiles per batch
  const int ctile = blockIdx.x & 7;
  const float* __restrict__ p = xyz + (size_t)b * (N_ * 3);

  // ---- async AoS -> SoA stage into LDS (ASYNCcnt path) ----
  const unsigned ldsbase = (unsigned)__builtin_amdgcn_groupstaticsize();
  for (int i = threadIdx.x; i < N_; i += 256) {
    const float* g = p + 3 * (size_t)i;
    const unsigned ox = ldsbase + 4u * (unsigned)i;
    const unsigned oy = ox + 4u * N_;
    const unsigned oz = oy + 4u * N_;
    asm volatile("global_load_async_to_lds_b32 %0, %1, off" :: "v"(ox), "v"(g)     : "memory");
    asm volatile("global_load_async_to_lds_b32 %0, %1, off" :: "v"(oy), "v"(g + 1) : "memory");
    asm volatile("global_load_async_to_lds_b32 %0, %1, off" :: "v"(oz), "v"(g + 2) : "memory");
  }
  asm volatile("s_wait_asynccnt 0" ::: "memory");
  __syncthreads();

  const int lane = threadIdx.x & 31;
  const int wv   = threadIdx.x >> 5;
  const int row  = lane & 15;                       // centroid row in 16-tile
  const int cidx = ctile * 128 + wv * 16 + row;     // centroid index in batch
  const float* cp = cent + ((size_t)b * M_ + cidx) * 3;
  const float cx = cp[0], cy = cp[1], cz = cp[2];
  const float cn = cx * cx + cy * cy + cz * cz;
  const bool lo  = (lane < 16);
  // A (16x4 f32): lanes 0-15 hold K0,K1 of row=lane; lanes 16-31 hold K2,K3.
  const v2f a = { lo ? -2.0f * cx : -2.0f * cz,
                  lo ? -2.0f * cy : cn };

  float* __restrict__ out = grouped + ((size_t)b * M_ + cidx) * (NS_ * 3);

  int cnt = lo ? 0 : NS_;                 // upper half never collects
  float fx = 0.f, fy = 0.f, fz = 0.f;     // first-hit relative coords (pad)
  const unsigned vsel = (unsigned)(lane & 7);

  for (int n0 = 0; n0 < N_; n0 += 16) {
    const int j = n0 + row;
    const float bxv = sx[j], byv = sy[j], bzv = sz[j];
    const float pn  = bxv * bxv + byv * byv + bzv * bzv;
    // B (4x16 f32, mirrored layout): lanes 0-15 K0,K1; lanes 16-31 K2,K3.
    const v2f bm = { lo ? bxv : bzv, lo ? byv : 1.0f };
    v8f c = { pn, pn, pn, pn, pn, pn, pn, pn };      // |p_n|^2 per column
    const v8f d = __builtin_amdgcn_wmma_f32_16x16x4_f32(
        false, a, false, bm, (short)0, c, false, false);

    // D layout: VGPR v, lanes 0-15 -> row v; lanes 16-31 -> row v+8.
    const unsigned b0 = (unsigned)__ballot(d[0] < R2_);
    const unsigned b1 = (unsigned)__ballot(d[1] < R2_);
    const unsigned b2 = (unsigned)__ballot(d[2] < R2_);
    const unsigned b3 = (unsigned)__ballot(d[3] < R2_);
    const unsigned b4 = (unsigned)__ballot(d[4] < R2_);
    const unsigned b5 = (unsigned)__ballot(d[5] < R2_);
    const unsigned b6 = (unsigned)__ballot(d[6] < R2_);
    const unsigned b7 = (unsigned)__ballot(d[7] < R2_);
    // lane r owns row r: ballot index r&7, half select by r&8
    const unsigned s0 = (vsel & 1u) ? b1 : b0;
    const unsigned s1 = (vsel & 1u) ? b3 : b2;
    const unsigned s2 = (vsel & 1u) ? b5 : b4;
    const unsigned s3 = (vsel & 1u) ? b7 : b6;
    const unsigned t0 = (vsel & 2u) ? s1 : s0;
    const unsigned t1 = (vsel & 2u) ? s3 : s2;
    const unsigned mw = (vsel & 4u) ? t1 : t0;
    unsigned mymask = (lane & 8) ? (mw >> 16) : (mw & 0xffffu);
    if (!lo) mymask = 0u;

    // pop hits in ascending index order (sparse: ~0.4% hit rate)
    while (mymask != 0u && cnt < NS_) {
      const int bit = __ffs(mymask) - 1;
      mymask &= mymask - 1u;
      const int idx = n0 + bit;
      const float hx = sx[idx] - cx;
      const float hy = sy[idx] - cy;
      const float hz = sz[idx] - cz;
      if (cnt == 0) { fx = hx; fy = hy; fz = hz; }
      float* o = out + cnt * 3;
      o[0] = hx; o[1] = hy; o[2] = hz;
      ++cnt;
    }
    if (__all(cnt >= NS_)) break;         // uniform early exit
  }

  if (lo) {                               // pad short lists with first hit
    for (int k = cnt; k < NS_; ++k) {
      float* o = out + k * 3;
      o[0] = fx; o[1] = fy; o[2] = fz;
    }
  }
}

// =============================== launcher ====================================
extern "C" void kernel_launch(void* const* d_in, const int* in_sizes, int n_in,
                              void* d_out, int out_size, void* d_ws, size_t ws_size,
                              hipStream_t stream) {
  (void)in_sizes; (void)n_in; (void)out_size; (void)d_ws; (void)ws_size;
  const float* xyz = (const float*)d_in[0];
  float* grouped = (float*)d_out;                           // [B,M,NS,3]
  float* centroids = grouped + (size_t)B_ * M_ * NS_ * 3;   // [B,M,3]

  fps_kernel<<<dim3(B_), dim3(1024), 0, stream>>>(xyz, centroids);

  (void)hipFuncSetAttribute((const void*)ballq_kernel,
                            hipFuncAttributeMaxDynamicSharedMemorySize,
                            (int)(3 * N_ * sizeof(float)));
  ballq_kernel<<<dim3(B_ * (M_ / 128)), dim3(256),
                 3 * N_ * sizeof(float), stream>>>(xyz, centroids, grouped);
}